// EFB_76510547411556
// MI455X (gfx1250) — compile-verified
//
#include <hip/hip_runtime.h>
#include <math.h>

// ---------------- problem constants ----------------
#define B_   4
#define C_   128
#define HC_  64
#define N_   65536          // H*W per batch
#define TN   64             // columns per block tile
#define LDT  66             // LDS tile stride (even -> 8B-aligned float2)
#define LDM  130            // merge tile stride (128 ch + pad, even)
#define EPSF 1e-5f
#define SCALEF (1.0f / (65536.0f * 64.0f))   // 1/(n*hc)

typedef float v2f __attribute__((ext_vector_type(2)));
typedef float v8f __attribute__((ext_vector_type(8)));

// ---------------- WMMA helpers (CDNA5, wave32) ----------------
// V_WMMA_F32_16X16X4_F32: A = 16x4 f32 (2 VGPR), B = 4x16 f32 (2 VGPR), C/D = 16x16 f32 (8 VGPR)
__device__ __forceinline__ v8f wmma4(v2f a, v2f b, v8f c) {
  return __builtin_amdgcn_wmma_f32_16x16x4_f32(
      /*neg_a=*/false, a, /*neg_b=*/false, b,
      /*c_mod=*/(short)0, c, /*reuse_a=*/false, /*reuse_b=*/false);
}

__device__ __forceinline__ v8f zero8() {
  v8f z;
#pragma unroll
  for (int i = 0; i < 8; ++i) z[i] = 0.0f;
  return z;
}

// A-frag / B-frag loader. Matrix stored row-major with the 16-wide dim as rows
// and the contraction (K) dim contiguous with row stride `ld` (even).
// Per ISA layout: lane's row = base + (lane&15); its two K slots = kk + 2*(lane>>4), +1.
__device__ __forceinline__ v2f ldfrag(const float* p, int row16, int ld, int kk, int lane) {
  const int r = row16 + (lane & 15);
  const int k = kk + ((lane >> 4) << 1);
  const float2 t = *reinterpret_cast<const float2*>(p + (size_t)r * ld + k);
  v2f o; o.x = t.x; o.y = t.y; return o;
}

// D-frag (16x16 f32): element (M = r + 8*(lane>>4), N = lane&15) in reg r.
__device__ __forceinline__ void stD_rowcol(float* p, int row16, int col16, int ld, int lane, v8f acc) {
  const int nd = lane & 15, hi = lane >> 4;
#pragma unroll
  for (int r = 0; r < 8; ++r)
    p[(size_t)(row16 + hi * 8 + r) * ld + col16 + nd] = acc[r];
}
__device__ __forceinline__ void stD_colrow(float* p, int row16, int col16, int ld, int lane, v8f acc) {
  const int nd = lane & 15, hi = lane >> 4;
#pragma unroll
  for (int r = 0; r < 8; ++r)
    p[(size_t)(col16 + nd) * ld + row16 + hi * 8 + r] = acc[r];
}

// ---------------- CDNA5 async global->LDS tile copy (transposing, dword granular) --------
// GLOBAL_LOAD_ASYNC_TO_LDS_B32: copies one dword per lane from global memory to LDS; tracked by ASYNCcnt.
// LDS operand is the low 32 bits of the flat pointer (LDS aperture truncation, ISA 10.2).
__device__ __forceinline__ void async_copy_b32(const float* gsrc, float* ldst) {
  const unsigned int  lds = (unsigned int)(unsigned long long)ldst;
  const unsigned long long ga = (unsigned long long)gsrc;
  asm volatile("global_load_async_to_lds_b32 %0, %1, off" :: "v"(lds), "v"(ga) : "memory");
}
__device__ __forceinline__ void wait_async_all() {
  asm volatile("s_wait_asynccnt 0x0" ::: "memory");
}
// Stage a 64ch x 64col tile (global row stride N_) into LDS as [col][ch] with stride LDT.
__device__ __forceinline__ void async_tile_load(float* dstTile, const float* gsrc, int tid) {
#pragma unroll 4
  for (int i = tid; i < HC_ * TN; i += 128) {
    const int ch = i >> 6, col = i & 63;
    async_copy_b32(gsrc + (size_t)ch * N_ + col, dstTile + (size_t)col * LDT + ch);
  }
}

// ---------------- kernel 0: zero the logit accumulators ----------------
__global__ void __launch_bounds__(256) kzero(float* p, int n) {
  int i = blockIdx.x * blockDim.x + threadIdx.x;
  if (i < n) p[i] = 0.0f;
}

// ---------------- kernel A: accumulate qk[b][i][j] = sum_n q[i,n]*k[j,n] ----------------
// grid = B * (N/TN) = 4096 blocks of 128 threads (4 waves). Wave w owns 16-col tile w.
__global__ void __launch_bounds__(128) kA(const float* __restrict__ x,
                                          const float* __restrict__ ln2_w,
                                          float* __restrict__ qk) {
  extern __shared__ float sm[];
  float* x2s = sm;                 // [col][ch]  64*66
  float* qs  = x2s + TN * LDT;     // [ch][col]  64*66
  float* ks  = qs  + HC_ * LDT;    // [ch][col]  64*66

  const int blk = blockIdx.x;
  const int b   = blk >> 10;            // /1024
  const int n0  = (blk & 1023) * TN;
  const float* x2g = x + ((size_t)(b * C_ + HC_)) * N_ + n0;

  async_tile_load(x2s, x2g, threadIdx.x);
  wait_async_all();
  __syncthreads();

  const int lane = threadIdx.x & 31, wave = threadIdx.x >> 5;
  const int colb = wave * 16;

  // q = ln2_w[0:64] @ x2 ; k = ln2_w[64:128] @ x2  (per wave: its 16 cols, all 4 M tiles)
#pragma unroll
  for (int part = 0; part < 2; ++part) {
    const float* W = ln2_w + part * HC_ * HC_;
    float* dst = part ? ks : qs;
    for (int mo = 0; mo < 4; ++mo) {
      v8f acc = zero8();
      for (int kk = 0; kk < HC_; kk += 4)
        acc = wmma4(ldfrag(W, mo * 16, HC_, kk, lane),
                    ldfrag(x2s, colb, LDT, kk, lane), acc);
      stD_rowcol(dst, mo * 16, colb, LDT, lane, acc);
    }
  }
  __syncthreads();

  // qk partial: contract this block's 64 columns; wave w owns j-tile w.
  float* qkb = qk + b * HC_ * HC_;
  for (int mo = 0; mo < 4; ++mo) {
    v8f acc = zero8();
    for (int kk = 0; kk < TN; kk += 4)
      acc = wmma4(ldfrag(qs, mo * 16, LDT, kk, lane),
                  ldfrag(ks, colb, LDT, kk, lane), acc);
    const int nd = lane & 15, hi = lane >> 4;
#pragma unroll
    for (int r = 0; r < 8; ++r)
      atomicAdd(&qkb[(mo * 16 + hi * 8 + r) * HC_ + colb + nd], acc[r]);
  }
}

// ---------------- softmax over rows of a [B*64 x 64] logit matrix ----------------
__global__ void __launch_bounds__(64) ksoftmax(const float* __restrict__ s,
                                               float* __restrict__ d) {
  __shared__ float red[64];
  const int row = blockIdx.x, j = threadIdx.x;
  const float v = s[row * 64 + j] * SCALEF;
  red[j] = v; __syncthreads();
  for (int off = 32; off > 0; off >>= 1) {
    if (j < off) red[j] = fmaxf(red[j], red[j + off]);
    __syncthreads();
  }
  const float m = red[0]; __syncthreads();
  const float e = __expf(v - m);
  red[j] = e; __syncthreads();
  for (int off = 32; off > 0; off >>= 1) {
    if (j < off) red[j] += red[j + off];
    __syncthreads();
  }
  d[row * 64 + j] = e / red[0];
}

// ---------------- kernel C: per tile compute nd1, v, a2, y2; accumulate cqk ----------------
__global__ void __launch_bounds__(128) kC(const float* __restrict__ x,
                                          const float* __restrict__ ln1_w,
                                          const float* __restrict__ ln2_w,
                                          const float* __restrict__ out2_w,
                                          const float* __restrict__ out2_g,
                                          const float* __restrict__ out2_b,
                                          const float* __restrict__ attn,
                                          float* __restrict__ cqk) {
  extern __shared__ float sm[];
  float* x1s  = sm;                 // [col][ch]
  float* x2s  = x1s  + TN * LDT;    // [col][ch]
  float* nd1s = x2s  + TN * LDT;    // [ch][col]
  float* vs   = nd1s + HC_ * LDT;   // [col][ch]
  float* a2s  = vs   + TN * LDT;    // [col][ch]
  float* y2s  = a2s  + TN * LDT;    // [ch][col]

  const int blk = blockIdx.x, b = blk >> 10, n0 = (blk & 1023) * TN;
  const float* x1g = x + (size_t)b * C_ * N_ + n0;
  const float* x2g = x1g + (size_t)HC_ * N_;
  async_tile_load(x1s, x1g, threadIdx.x);
  async_tile_load(x2s, x2g, threadIdx.x);
  wait_async_all();
  __syncthreads();

  const int lane = threadIdx.x & 31, wave = threadIdx.x >> 5;
  const int colb = wave * 16;
  const int nd = lane & 15, hi = lane >> 4;

  // nd1 = ln1_w @ x1 ; v = ln2_w[128:192] @ x2
  const float* Wv = ln2_w + 2 * HC_ * HC_;
  for (int mo = 0; mo < 4; ++mo) {
    v8f a0 = zero8(), a1 = zero8();
    for (int kk = 0; kk < HC_; kk += 4) {
      a0 = wmma4(ldfrag(ln1_w, mo * 16, HC_, kk, lane), ldfrag(x1s, colb, LDT, kk, lane), a0);
      a1 = wmma4(ldfrag(Wv,    mo * 16, HC_, kk, lane), ldfrag(x2s, colb, LDT, kk, lane), a1);
    }
    stD_rowcol(nd1s, mo * 16, colb, LDT, lane, a0);
    stD_colrow(vs,   mo * 16, colb, LDT, lane, a1);
  }
  __syncthreads();

  // a2 = attn @ v
  const float* attn_b = attn + b * HC_ * HC_;
  for (int mo = 0; mo < 4; ++mo) {
    v8f acc = zero8();
    for (int kk = 0; kk < HC_; kk += 4)
      acc = wmma4(ldfrag(attn_b, mo * 16, HC_, kk, lane), ldfrag(vs, colb, LDT, kk, lane), acc);
    stD_colrow(a2s, mo * 16, colb, LDT, lane, acc);
  }
  __syncthreads();

  // y2 = LN_ch(out2_w @ [x2; a2])
  v8f yacc[4];
  for (int mo = 0; mo < 4; ++mo) {
    v8f acc = zero8();
    for (int kk = 0; kk < C_; kk += 4) {
      v2f bf = (kk < HC_) ? ldfrag(x2s, colb, LDT, kk, lane)
                          : ldfrag(a2s, colb, LDT, kk - HC_, lane);
      acc = wmma4(ldfrag(out2_w, mo * 16, C_, kk, lane), bf, acc);
    }
    yacc[mo] = acc;
  }
  float s = 0.0f, ss = 0.0f;
#pragma unroll
  for (int mo = 0; mo < 4; ++mo)
#pragma unroll
    for (int r = 0; r < 8; ++r) { const float t = yacc[mo][r]; s += t; ss += t * t; }
  s  += __shfl_xor(s, 16);
  ss += __shfl_xor(ss, 16);
  const float mean = s * (1.0f / 64.0f);
  const float inv  = rsqrtf(ss * (1.0f / 64.0f) - mean * mean + EPSF);
#pragma unroll
  for (int mo = 0; mo < 4; ++mo)
#pragma unroll
    for (int r = 0; r < 8; ++r) {
      const int ch = mo * 16 + hi * 8 + r;
      y2s[(size_t)ch * LDT + colb + nd] =
          (yacc[mo][r] - mean) * inv * out2_g[ch] + out2_b[ch];
    }
  __syncthreads();

  // cqk partial: nd1 . y2 over this block's 64 columns; wave w owns j-tile w.
  float* cqk_b = cqk + b * HC_ * HC_;
  for (int mo = 0; mo < 4; ++mo) {
    v8f acc = zero8();
    for (int kk = 0; kk < TN; kk += 4)
      acc = wmma4(ldfrag(nd1s, mo * 16, LDT, kk, lane),
                  ldfrag(y2s, colb, LDT, kk, lane), acc);
#pragma unroll
    for (int r = 0; r < 8; ++r)
      atomicAdd(&cqk_b[(mo * 16 + hi * 8 + r) * HC_ + colb + nd], acc[r]);
  }
}

// ---------------- kernel E: recompute y2, gate x1, y1, final GEMM+LN+relu+residual ----------
__global__ void __launch_bounds__(128) kE(const float* __restrict__ x,
                                          const float* __restrict__ ln2_w,
                                          const float* __restrict__ out1_w,
                                          const float* __restrict__ out1_g,
                                          const float* __restrict__ out1_b,
                                          const float* __restrict__ out2_w,
                                          const float* __restrict__ out2_g,
                                          const float* __restrict__ out2_b,
                                          const float* __restrict__ out_w,
                                          const float* __restrict__ out_g,
                                          const float* __restrict__ out_b,
                                          const float* __restrict__ attn,
                                          const float* __restrict__ cattn,
                                          float* __restrict__ out) {
  extern __shared__ float sm[];
  float* x1s  = sm;                 // [col][ch]
  float* x2s  = x1s  + TN * LDT;    // [col][ch]
  float* vs   = x2s  + TN * LDT;    // [col][ch]
  float* a2s  = vs   + TN * LDT;    // [col][ch]
  float* x1gs = a2s  + TN * LDT;    // [col][ch]
  float* mrg  = x1gs + TN * LDT;    // [col][128] stride LDM; y1 at +0, y2 at +64

  const int blk = blockIdx.x, b = blk >> 10, n0 = (blk & 1023) * TN;
  const float* x1g = x + (size_t)b * C_ * N_ + n0;
  const float* x2g = x1g + (size_t)HC_ * N_;
  async_tile_load(x1s, x1g, threadIdx.x);
  async_tile_load(x2s, x2g, threadIdx.x);
  wait_async_all();
  __syncthreads();

  const int lane = threadIdx.x & 31, wave = threadIdx.x >> 5;
  const int colb = wave * 16;
  const int nd = lane & 15, hi = lane >> 4;

  // v = ln2_w[128:192] @ x2
  const float* Wv = ln2_w + 2 * HC_ * HC_;
  for (int mo = 0; mo < 4; ++mo) {
    v8f acc = zero8();
    for (int kk = 0; kk < HC_; kk += 4)
      acc = wmma4(ldfrag(Wv, mo * 16, HC_, kk, lane), ldfrag(x2s, colb, LDT, kk, lane), acc);
    stD_colrow(vs, mo * 16, colb, LDT, lane, acc);
  }
  __syncthreads();

  // a2 = attn @ v
  const float* attn_b = attn + b * HC_ * HC_;
  for (int mo = 0; mo < 4; ++mo) {
    v8f acc = zero8();
    for (int kk = 0; kk < HC_; kk += 4)
      acc = wmma4(ldfrag(attn_b, mo * 16, HC_, kk, lane), ldfrag(vs, colb, LDT, kk, lane), acc);
    stD_colrow(a2s, mo * 16, colb, LDT, lane, acc);
  }
  __syncthreads();

  // y2 = LN_ch(out2_w @ [x2; a2]) -> mrg[col][64+ch]
  {
    v8f yacc[4];
    for (int mo = 0; mo < 4; ++mo) {
      v8f acc = zero8();
      for (int kk = 0; kk < C_; kk += 4) {
        v2f bf = (kk < HC_) ? ldfrag(x2s, colb, LDT, kk, lane)
                            : ldfrag(a2s, colb, LDT, kk - HC_, lane);
        acc = wmma4(ldfrag(out2_w, mo * 16, C_, kk, lane), bf, acc);
      }
      yacc[mo] = acc;
    }
    float s = 0.0f, ss = 0.0f;
#pragma unroll
    for (int mo = 0; mo < 4; ++mo)
#pragma unroll
      for (int r = 0; r < 8; ++r) { const float t = yacc[mo][r]; s += t; ss += t * t; }
    s  += __shfl_xor(s, 16);
    ss += __shfl_xor(ss, 16);
    const float mean = s * (1.0f / 64.0f);
    const float inv  = rsqrtf(ss * (1.0f / 64.0f) - mean * mean + EPSF);
#pragma unroll
    for (int mo = 0; mo < 4; ++mo)
#pragma unroll
      for (int r = 0; r < 8; ++r) {
        const int ch = mo * 16 + hi * 8 + r;
        mrg[(size_t)(colb + nd) * LDM + HC_ + ch] =
            (yacc[mo][r] - mean) * inv * out2_g[ch] + out2_b[ch];
      }
  }
  __syncthreads();

  // corr = sigmoid(cattn @ y2); x1g = x1*corr + x1
  const float* cattn_b = cattn + b * HC_ * HC_;
  for (int mo = 0; mo < 4; ++mo) {
    v8f acc = zero8();
    for (int kk = 0; kk < HC_; kk += 4)
      acc = wmma4(ldfrag(cattn_b, mo * 16, HC_, kk, lane),
                  ldfrag(mrg + HC_, colb, LDM, kk, lane), acc);
#pragma unroll
    for (int r = 0; r < 8; ++r) {
      const int ch = mo * 16 + hi * 8 + r;
      const int col = colb + nd;
      const float cv = 1.0f / (1.0f + __expf(-acc[r]));
      const float xv = x1s[(size_t)col * LDT + ch];
      x1gs[(size_t)col * LDT + ch] = xv * cv + xv;
    }
  }
  __syncthreads();

  // y1 = LN_ch(out1_w @ x1g) -> mrg[col][ch]
  {
    v8f yacc[4];
    for (int mo = 0; mo < 4; ++mo) {
      v8f acc = zero8();
      for (int kk = 0; kk < HC_; kk += 4)
        acc = wmma4(ldfrag(out1_w, mo * 16, HC_, kk, lane),
                    ldfrag(x1gs, colb, LDT, kk, lane), acc);
      yacc[mo] = acc;
    }
    float s = 0.0f, ss = 0.0f;
#pragma unroll
    for (int mo = 0; mo < 4; ++mo)
#pragma unroll
      for (int r = 0; r < 8; ++r) { const float t = yacc[mo][r]; s += t; ss += t * t; }
    s  += __shfl_xor(s, 16);
    ss += __shfl_xor(ss, 16);
    const float mean = s * (1.0f / 64.0f);
    const float inv  = rsqrtf(ss * (1.0f / 64.0f) - mean * mean + EPSF);
#pragma unroll
    for (int mo = 0; mo < 4; ++mo)
#pragma unroll
      for (int r = 0; r < 8; ++r) {
        const int ch = mo * 16 + hi * 8 + r;
        mrg[(size_t)(colb + nd) * LDM + ch] =
            (yacc[mo][r] - mean) * inv * out1_g[ch] + out1_b[ch];
      }
  }
  __syncthreads();

  // out = relu(LN_ch(out_w @ merge)) + x
  v8f oacc[8];
  for (int mo = 0; mo < 8; ++mo) {
    v8f acc = zero8();
    for (int kk = 0; kk < C_; kk += 4)
      acc = wmma4(ldfrag(out_w, mo * 16, C_, kk, lane),
                  ldfrag(mrg, colb, LDM, kk, lane), acc);
    oacc[mo] = acc;
  }
  float s = 0.0f, ss = 0.0f;
#pragma unroll
  for (int mo = 0; mo < 8; ++mo)
#pragma unroll
    for (int r = 0; r < 8; ++r) { const float t = oacc[mo][r]; s += t; ss += t * t; }
  s  += __shfl_xor(s, 16);
  ss += __shfl_xor(ss, 16);
  const float mean = s * (1.0f / 128.0f);
  const float inv  = rsqrtf(ss * (1.0f / 128.0f) - mean * mean + EPSF);
  const int col = colb + nd;
#pragma unroll
  for (int mo = 0; mo < 8; ++mo)
#pragma unroll
    for (int r = 0; r < 8; ++r) {
      const int ch = mo * 16 + hi * 8 + r;
      float v = (oacc[mo][r] - mean) * inv * out_g[ch] + out_b[ch];
      v = fmaxf(v, 0.0f);
      const float resid = (ch < HC_) ? x1s[(size_t)col * LDT + ch]
                                     : x2s[(size_t)col * LDT + (ch - HC_)];
      out[(size_t)(b * C_ + ch) * N_ + n0 + col] = v + resid;
    }
}

// ---------------- host launcher ----------------
extern "C" void kernel_launch(void* const* d_in, const int* in_sizes, int n_in,
                              void* d_out, int out_size, void* d_ws, size_t ws_size,
                              hipStream_t stream) {
  const float* x      = (const float*)d_in[0];
  const float* ln1_w  = (const float*)d_in[1];
  const float* ln2_w  = (const float*)d_in[2];
  const float* out1_w = (const float*)d_in[3];
  const float* out1_g = (const float*)d_in[4];
  const float* out1_b = (const float*)d_in[5];
  const float* out2_w = (const float*)d_in[6];
  const float* out2_g = (const float*)d_in[7];
  const float* out2_b = (const float*)d_in[8];
  const float* out_w  = (const float*)d_in[9];
  const float* out_g  = (const float*)d_in[10];
  const float* out_b  = (const float*)d_in[11];
  float* out = (float*)d_out;

  float* qk    = (float*)d_ws;          // [B][64][64]
  float* cqk   = qk    + B_ * HC_ * HC_;
  float* attn  = cqk   + B_ * HC_ * HC_;
  float* cattn = attn  + B_ * HC_ * HC_;

  const int nlogit = 2 * B_ * HC_ * HC_;          // zero qk + cqk
  kzero<<<(nlogit + 255) / 256, 256, 0, stream>>>(qk, nlogit);

  const dim3 grid(B_ * (N_ / TN));                 // 4096 tile blocks
  const dim3 blk(128);                             // 4 waves

  const size_t smA = (size_t)(3 * TN * LDT) * sizeof(float);                  //  ~50 KB
  const size_t smC = (size_t)(6 * TN * LDT) * sizeof(float);                  // ~101 KB
  const size_t smE = (size_t)(5 * TN * LDT + TN * LDM) * sizeof(float);       // ~118 KB

  kA<<<grid, blk, smA, stream>>>(x, ln2_w, qk);
  ksoftmax<<<B_ * HC_, 64, 0, stream>>>(qk, attn);
  kC<<<grid, blk, smC, stream>>>(x, ln1_w, ln2_w, out2_w, out2_g, out2_b, attn, cqk);
  ksoftmax<<<B_ * HC_, 64, 0, stream>>>(cqk, cattn);
  kE<<<grid, blk, smE, stream>>>(x, ln2_w, out1_w, out1_g, out1_b,
                                 out2_w, out2_g, out2_b,
                                 out_w, out_g, out_b, attn, cattn, out);
}